// MultiHeadRelativeAttention_72799695667785
// MI455X (gfx1250) — compile-verified
//
#include <hip/hip_runtime.h>
#include <hip/hip_bf16.h>

// MultiHeadRelativeAttention on gfx1250 (MI455X), fp32 WMMA throughout.
// B=2, H=8, S=2048, hd=32, D=256, NUM_RPRS=2049, MAX_REL=1024.
// 3 kernels: QKV projection -> fused attention (logits+softmax+AV) -> out proj.

#define BATCH   2
#define NHEADS  8
#define SEQ     2048
#define DIMS    256
#define HD      32
#define NRPR    2049

typedef __attribute__((ext_vector_type(2))) float v2f;
typedef __attribute__((ext_vector_type(8))) float v8f;

__device__ __forceinline__ v8f wmma4(v2f a, v2f b, v8f c) {
    // D = A(16x4, f32) * B(4x16, f32) + C(16x16, f32)
    return __builtin_amdgcn_wmma_f32_16x16x4_f32(false, a, false, b, (short)0, c,
                                                 false, false);
}

__device__ __forceinline__ int clamp_rpr(int d) {
    return d < 0 ? 0 : (d > (NRPR - 1) ? (NRPR - 1) : d);
}

// ---------------------------------------------------------------------------
// Kernel 1: fused Q/K/V projection.  X(4096x256) @ W(256x256) + b
// grid = (M/16, 3), block = 128 (4 waves).  Each wave: 16x64 output strip.
// Output stored per-head: [B][H][S][HD].
// ---------------------------------------------------------------------------
__global__ void qkv_proj_kernel(const float* __restrict__ query,
                                const float* __restrict__ value,
                                const float* __restrict__ Wq, const float* __restrict__ bq,
                                const float* __restrict__ Wk, const float* __restrict__ bk,
                                const float* __restrict__ Wv, const float* __restrict__ bv,
                                float* __restrict__ Qh, float* __restrict__ Kh,
                                float* __restrict__ Vh)
{
    const int p = blockIdx.y;
    const float* X    = (p == 0) ? query : value;
    const float* W    = (p == 0) ? Wq : (p == 1) ? Wk : Wv;
    const float* bias = (p == 0) ? bq : (p == 1) ? bk : bv;
    float*       dst  = (p == 0) ? Qh : (p == 1) ? Kh : Vh;

    const int lane = threadIdx.x & 31;
    const int wave = threadIdx.x >> 5;
    const int ln   = lane & 15;
    const int hw   = lane >> 4;
    const int k2   = hw * 2;
    const int row0 = blockIdx.x * 16;
    const int col0 = wave * 64;

    v8f acc[4] = {};
    const float* Arow = X + (size_t)(row0 + ln) * DIMS;
    for (int k = 0; k < DIMS; k += 4) {
        v2f a; a.x = Arow[k + k2]; a.y = Arow[k + k2 + 1];
        const float* W0 = W + (size_t)(k + k2) * DIMS + col0 + ln;
#pragma unroll
        for (int t = 0; t < 4; ++t) {
            v2f b; b.x = W0[t * 16]; b.y = W0[t * 16 + DIMS];
            acc[t] = wmma4(a, b, acc[t]);
        }
    }
#pragma unroll
    for (int t = 0; t < 4; ++t) {
        const int col = col0 + t * 16 + ln;
        const int h   = col >> 5;
        const int e   = col & (HD - 1);
        const float bb = bias[col];
#pragma unroll
        for (int i = 0; i < 8; ++i) {
            const int m    = i + 8 * hw;
            const int row  = row0 + m;                // b*SEQ + s
            const int bidx = row >> 11;
            const int s    = row & (SEQ - 1);
            dst[(((size_t)bidx * NHEADS + h) * SEQ + s) * HD + e] = acc[t][i] + bb;
        }
    }
}

// ---------------------------------------------------------------------------
// Kernel 2: fused attention.  One block = one (b,h,q-tile of 16) over all
// 2048 keys.  8 waves x 256 keys each; logits live in registers (16 x v8f).
// Phase A: logits = (Q K^T + Q E^T)/sqrt(hd)   (content + windowed-P WMMA)
// Phase B: block softmax via LDS reductions
// Phase C: write weights to d_out once, and compute W@V + sheared(W)@T_win
//          into 16x32 accumulators; 8-wave LDS reduction -> Oh.
// ---------------------------------------------------------------------------
__global__ void __launch_bounds__(256) attention_kernel(
    const float* __restrict__ Qh, const float* __restrict__ Kh,
    const float* __restrict__ Vh, const float* __restrict__ rpr,
    float* __restrict__ attn, float* __restrict__ Oh)
{
    __shared__ float smem[8 * 16 * 80];   // per-wave 16x80 tile; Ored overlay at end
    __shared__ float red[16][8][16];      // per-query partial reductions
    __shared__ float rowmax_s[16];
    __shared__ float rowsum_s[16];

    const int tid  = threadIdx.x;
    const int lane = tid & 31;
    const int wave = tid >> 5;            // 0..7
    const int ln   = lane & 15;
    const int hw   = lane >> 4;
    const int k2   = hw * 2;
    const int q0   = blockIdx.x * 16;
    const int bh   = blockIdx.z;
    const int b    = bh >> 3;
    const int h    = bh & 7;

    const float* Q = Qh + (size_t)bh * SEQ * HD;
    const float* K = Kh + (size_t)bh * SEQ * HD;
    const float* V = Vh + (size_t)bh * SEQ * HD;
    float* Wtile = smem + wave * (16 * 80);   // this wave's [16][80] tile
    const int kwave0 = wave * 256;            // this wave's key range

    const float scale = 0.17677669529663687f; // 1/sqrt(32)

    // ---------------- Phase A: logits into registers ----------------
    v8f larr[16] = {};                        // [chunk*4 + t], 16x64 per chunk
    v2f aq[8];                                // Q fragments, reused across chunks
    {
        const float* Aq = Q + (size_t)(q0 + ln) * HD;
#pragma unroll
        for (int ks = 0; ks < 8; ++ks) {
            aq[ks].x = Aq[ks * 4 + k2];
            aq[ks].y = Aq[ks * 4 + k2 + 1];
        }
    }
#pragma unroll
    for (int chunk = 0; chunk < 4; ++chunk) {
        const int kc = kwave0 + chunk * 64;
        const int dbase = kc - q0 - 15 + 1024;
        v8f pacc[5] = {};
        for (int ks = 0; ks < 8; ++ks) {
            const v2f a = aq[ks];
            const int kk = ks * 4 + k2;
#pragma unroll
            for (int t = 0; t < 4; ++t) {
                const float* Kr = K + (size_t)(kc + t * 16 + ln) * HD + kk;
                v2f bf; bf.x = Kr[0]; bf.y = Kr[1];
                larr[chunk * 4 + t] = wmma4(a, bf, larr[chunk * 4 + t]);
            }
#pragma unroll
            for (int t = 0; t < 5; ++t) {
                const int d = clamp_rpr(dbase + t * 16 + ln);
                const float* Tr = rpr + (size_t)d * HD + kk;
                v2f bf; bf.x = Tr[0]; bf.y = Tr[1];
                pacc[t] = wmma4(a, bf, pacc[t]);
            }
        }
        // gather P via wave-local LDS (in-order DS ops; no barrier needed)
#pragma unroll
        for (int t = 0; t < 5; ++t)
#pragma unroll
            for (int i = 0; i < 8; ++i)
                Wtile[(i + 8 * hw) * 80 + t * 16 + ln] = pacc[t][i];
#pragma unroll
        for (int t = 0; t < 4; ++t)
#pragma unroll
            for (int i = 0; i < 8; ++i) {
                const int m = i + 8 * hw;
                larr[chunk * 4 + t][i] =
                    (larr[chunk * 4 + t][i] + Wtile[m * 80 + t * 16 + ln + 15 - m]) * scale;
            }
    }

    // ---------------- Phase B: block softmax ----------------
    {
        float pm[8];
#pragma unroll
        for (int i = 0; i < 8; ++i) pm[i] = -3.0e38f;
#pragma unroll
        for (int j = 0; j < 16; ++j)
#pragma unroll
            for (int i = 0; i < 8; ++i) pm[i] = fmaxf(pm[i], larr[j][i]);
#pragma unroll
        for (int i = 0; i < 8; ++i) red[i + 8 * hw][wave][ln] = pm[i];
    }
    __syncthreads();
    if (tid < 16) {
        float mx = -3.0e38f;
        const float* r = &red[tid][0][0];
        for (int j = 0; j < 128; ++j) mx = fmaxf(mx, r[j]);
        rowmax_s[tid] = mx;
    }
    __syncthreads();
    {
        float mx8[8], ps[8] = {};
#pragma unroll
        for (int i = 0; i < 8; ++i) mx8[i] = rowmax_s[i + 8 * hw];
#pragma unroll
        for (int j = 0; j < 16; ++j)
#pragma unroll
            for (int i = 0; i < 8; ++i) {
                const float e = expf(larr[j][i] - mx8[i]);
                larr[j][i] = e;
                ps[i] += e;
            }
#pragma unroll
        for (int i = 0; i < 8; ++i) red[i + 8 * hw][wave][ln] = ps[i];
    }
    __syncthreads();
    if (tid < 16) {
        float s = 0.f;
        const float* r = &red[tid][0][0];
        for (int j = 0; j < 128; ++j) s += r[j];
        rowsum_s[tid] = 1.f / s;
    }
    __syncthreads();
    {
        float inv8[8];
#pragma unroll
        for (int i = 0; i < 8; ++i) inv8[i] = rowsum_s[i + 8 * hw];
#pragma unroll
        for (int j = 0; j < 16; ++j)
#pragma unroll
            for (int i = 0; i < 8; ++i) larr[j][i] *= inv8[i];
    }

    // ---------------- Phase C: emit weights + aggregate values ----------------
    // zero Wtile once; sheared band overwrites the same cells every chunk
    for (int j = lane; j < 16 * 80; j += 32) Wtile[j] = 0.f;

    v8f acc[2] = {};
#pragma unroll
    for (int chunk = 0; chunk < 4; ++chunk) {
        const int kc = kwave0 + chunk * 64;
        const int dbase = kc - q0 - 15 + 1024;
        // scatter normalized weights: global write (only touch of attn buffer)
        // + sheared LDS tile W2[m][kloc + 15 - m]
#pragma unroll
        for (int t = 0; t < 4; ++t)
#pragma unroll
            for (int i = 0; i < 8; ++i) {
                const int m = i + 8 * hw;
                const float w = larr[chunk * 4 + t][i];
                attn[((size_t)bh * SEQ + q0 + m) * SEQ + kc + t * 16 + ln] = w;
                Wtile[m * 80 + (t * 16 + ln) + 15 - m] = w;
            }
        // content GEMM: A(16x64) read un-sheared, B = V(64x32)
        for (int ks = 0; ks < 16; ++ks) {
            const int kk = ks * 4 + k2;
            v2f a;
            a.x = Wtile[ln * 80 + kk + 15 - ln];
            a.y = Wtile[ln * 80 + kk + 16 - ln];
#pragma unroll
            for (int t = 0; t < 2; ++t) {
                const float* Vr = V + (size_t)(kc + kk) * HD + t * 16 + ln;
                v2f bf; bf.x = Vr[0]; bf.y = Vr[HD];
                acc[t] = wmma4(a, bf, acc[t]);
            }
        }
        // relative GEMM: A = W2(16x80), B = T_win(80x32)
        for (int cs = 0; cs < 20; ++cs) {
            const int c = cs * 4 + k2;
            v2f a;
            a.x = Wtile[ln * 80 + c];
            a.y = Wtile[ln * 80 + c + 1];
            const int dA = clamp_rpr(dbase + c);
            const int dB = clamp_rpr(dbase + c + 1);
#pragma unroll
            for (int t = 0; t < 2; ++t) {
                v2f bf;
                bf.x = rpr[(size_t)dA * HD + t * 16 + ln];
                bf.y = rpr[(size_t)dB * HD + t * 16 + ln];
                acc[t] = wmma4(a, bf, acc[t]);
            }
        }
    }

    __syncthreads();                       // before overlaying smem with Ored
    float* Ored = smem;                    // [8][16][32]
#pragma unroll
    for (int t = 0; t < 2; ++t)
#pragma unroll
        for (int i = 0; i < 8; ++i)
            Ored[(wave * 16 + i + 8 * hw) * 32 + t * 16 + ln] = acc[t][i];
    __syncthreads();
    for (int idx = tid; idx < 16 * 32; idx += 256) {
        const int m = idx >> 5;
        const int e = idx & 31;
        float s = 0.f;
#pragma unroll
        for (int w2 = 0; w2 < 8; ++w2) s += Ored[(w2 * 16 + m) * 32 + e];
        Oh[((size_t)b * SEQ + q0 + m) * DIMS + h * HD + e] = s;
    }
}

// ---------------------------------------------------------------------------
// Kernel 3: out = Oh(4096x256) @ Wo(256x256) + bo  -> d_out[0 : B*S*D]
// ---------------------------------------------------------------------------
__global__ void outproj_kernel(const float* __restrict__ Oh,
                               const float* __restrict__ Wo,
                               const float* __restrict__ bo,
                               float* __restrict__ out)
{
    const int lane = threadIdx.x & 31;
    const int wave = threadIdx.x >> 5;
    const int ln   = lane & 15;
    const int hw   = lane >> 4;
    const int k2   = hw * 2;
    const int row0 = blockIdx.x * 16;
    const int col0 = wave * 64;

    v8f acc[4] = {};
    const float* Arow = Oh + (size_t)(row0 + ln) * DIMS;
    for (int k = 0; k < DIMS; k += 4) {
        v2f a; a.x = Arow[k + k2]; a.y = Arow[k + k2 + 1];
        const float* W0 = Wo + (size_t)(k + k2) * DIMS + col0 + ln;
#pragma unroll
        for (int t = 0; t < 4; ++t) {
            v2f b; b.x = W0[t * 16]; b.y = W0[t * 16 + DIMS];
            acc[t] = wmma4(a, b, acc[t]);
        }
    }
#pragma unroll
    for (int t = 0; t < 4; ++t) {
        const int col = col0 + t * 16 + ln;
        const float bb = bo[col];
#pragma unroll
        for (int i = 0; i < 8; ++i) {
            const int m = i + 8 * hw;
            out[(size_t)(row0 + m) * DIMS + col] = acc[t][i] + bb;
        }
    }
}

extern "C" void kernel_launch(void* const* d_in, const int* in_sizes, int n_in,
                              void* d_out, int out_size, void* d_ws, size_t ws_size,
                              hipStream_t stream)
{
    (void)in_sizes; (void)n_in; (void)out_size; (void)ws_size;

    const float* query = (const float*)d_in[0];
    const float* value = (const float*)d_in[1];
    const float* Wq = (const float*)d_in[2];
    const float* bq = (const float*)d_in[3];
    const float* Wk = (const float*)d_in[4];
    const float* bk = (const float*)d_in[5];
    const float* Wv = (const float*)d_in[6];
    const float* bv = (const float*)d_in[7];
    const float* Wo = (const float*)d_in[8];
    const float* bo = (const float*)d_in[9];
    const float* rpr = (const float*)d_in[10];
    // d_in[11] (rpr_lookup) unused: index = clip(k-q,-1024,1024)+1024 computed exactly.

    float* out  = (float*)d_out;
    float* attn = out + (size_t)BATCH * SEQ * DIMS;   // (B,H,S,S) region of d_out

    const size_t headsz = (size_t)BATCH * NHEADS * SEQ * HD;  // 1,048,576 floats
    float* Qh = (float*)d_ws;
    float* Kh = Qh + headsz;
    float* Vh = Kh + headsz;
    float* Oh = Vh + headsz;                          // (B,S,D) pre-projection

    qkv_proj_kernel<<<dim3((BATCH * SEQ) / 16, 3, 1), 128, 0, stream>>>(
        query, value, Wq, bq, Wk, bk, Wv, bv, Qh, Kh, Vh);
    attention_kernel<<<dim3(SEQ / 16, 1, BATCH * NHEADS), 256, 0, stream>>>(
        Qh, Kh, Vh, rpr, attn, Oh);
    outproj_kernel<<<dim3((BATCH * SEQ) / 16, 1, 1), 128, 0, stream>>>(
        Oh, Wo, bo, out);
}